// Conv2d_res_v3_58978490908851
// MI455X (gfx1250) — compile-verified
//
#include <hip/hip_runtime.h>
#include <hip/hip_bf16.h>
#include <math.h>

#ifndef __has_builtin
#define __has_builtin(x) 0
#endif

#define GAS __attribute__((address_space(1)))
#define LAS __attribute__((address_space(3)))

// Problem geometry (fixed by the reference)
constexpr int H = 224, W = 224, C = 3;
constexpr int TILE_H = 16;                // rows of output per block
constexpr int TILES  = H / TILE_H;        // 14 tiles per plane
constexpr float KA = 0.35355339059327373f; // 1/(2*sqrt(2))

// One block: 224 threads (7 wave32), one 16-row strip of one (n,c) plane.
// LDS tile: (TILE_H+2) x (W+2) with zero halo implementing padding=1.
// ~16.3 KB LDS out of 320 KB/WGP -> occupancy is launch-bound, not LDS-bound.
__global__ __launch_bounds__(W) void sobel_gabor_fused_kernel(
    const float* __restrict__ x, const float* __restrict__ mask,
    float* __restrict__ out, int nBatch)
{
    __shared__ float smem[TILE_H + 2][W + 2];

    const int tile = blockIdx.x % TILES;
    const int p    = blockIdx.x / TILES;   // plane index = n*C + c
    const int c    = p % C;
    const int n    = p / C;
    const int w    = threadIdx.x;          // 0..223 (column)
    const int h0   = tile * TILE_H;

    const size_t plane = (size_t)H * W;
    const float* __restrict__ xp = x    + (size_t)p * plane;
    const float* __restrict__ mp = mask + (size_t)c * plane;
    float* __restrict__ normp = out + ((size_t)n * 6 + c)     * plane;
    float* __restrict__ gabp  = out + ((size_t)n * 6 + 3 + c) * plane;
    float* __restrict__ copyp = out + (size_t)nBatch * 6 * plane + (size_t)p * plane;

    // Warm the mask rows through the cache while the tile streams to LDS.
    __builtin_prefetch(mp + (size_t)h0 * W + w, 0, 1);

    // Zero the left/right halo columns (threads 0 and 1).
    if (w < 2) {
        #pragma unroll
        for (int r = 0; r < TILE_H + 2; ++r)
            smem[r][w * (W + 1)] = 0.0f;   // col 0 or col W+1
    }

    // Stage (TILE_H+2) rows x W columns into LDS interior via gfx1250
    // async global->LDS DMA (no VGPR staging, ASYNCcnt-tracked).
    // Branch on h is uniform across the block (depends only on r).
    #pragma unroll
    for (int r = 0; r < TILE_H + 2; ++r) {
        const int h = h0 - 1 + r;
        if (h >= 0 && h < H) {
#if __has_builtin(__builtin_amdgcn_global_load_async_to_lds_b32)
            __builtin_amdgcn_global_load_async_to_lds_b32(
                (GAS int*)(xp + (size_t)h * W + w),
                (LAS int*)&smem[r][w + 1],
                /*offset=*/0, /*cpol=*/0);
#else
            smem[r][w + 1] = xp[(size_t)h * W + w];
#endif
        } else {
            smem[r][w + 1] = 0.0f;         // zero padding rows
        }
    }

#if __has_builtin(__builtin_amdgcn_global_load_async_to_lds_b32)
#if __has_builtin(__builtin_amdgcn_s_wait_asynccnt)
    __builtin_amdgcn_s_wait_asynccnt(0);
#else
    asm volatile("s_wait_asynccnt 0" ::: "memory");
#endif
#endif
    __syncthreads();

    // Compute TILE_H output rows; 9 LDS reads per pixel, conflict-free
    // (consecutive lanes touch consecutive banks within a row).
    #pragma unroll
    for (int r = 0; r < TILE_H; ++r) {
        const int h = h0 + r;

        const float t00 = smem[r    ][w], t01 = smem[r    ][w + 1], t02 = smem[r    ][w + 2];
        const float t10 = smem[r + 1][w], xc  = smem[r + 1][w + 1], t12 = smem[r + 1][w + 2];
        const float t20 = smem[r + 2][w], t21 = smem[r + 2][w + 1], t22 = smem[r + 2][w + 2];

        // Cross-correlation with KX / KY (no kernel flip in XLA conv):
        // KX = [[-A,0,A],[-1,0,1],[-A,0,A]], KY = [[A,1,A],[0,0,0],[-A,-1,-A]]
        const float gx = KA * (t02 - t00) + (t12 - t10) + KA * (t22 - t20);
        const float gy = KA * (t00 - t20) + (t01 - t21) + KA * (t02 - t22);

        // (gx*inv)^2 + (gy*inv)^2 == inv^2 * (gx^2 + gy^2)
        //  -> sqrt(...) == |inv| * sqrt(gx^2 + gy^2)
        const float inv = 1.0f / (xc + 0.001f);
        const float mag = sqrtf(gx * gx + gy * gy) * fabsf(inv);
        const float gab = atanf(mag);

        const size_t o = (size_t)h * W + w;
        // Outputs are write-once streams (231 MB > 192 MB L2): use NT stores
        // so L2 stays reserved for the x halo re-reads and the mask.
        __builtin_nontemporal_store(xc * mp[o], normp + o); // channel c   : x*mask
        __builtin_nontemporal_store(gab,        gabp  + o); // channel 3+c : gabor
        __builtin_nontemporal_store(xc,         copyp + o); // tuple elem 2: x copy
    }
}

extern "C" void kernel_launch(void* const* d_in, const int* in_sizes, int n_in,
                              void* d_out, int out_size, void* d_ws, size_t ws_size,
                              hipStream_t stream) {
    const float* x    = (const float*)d_in[0];
    const float* mask = (const float*)d_in[1];
    float* out        = (float*)d_out;

    const int nBatch = in_sizes[0] / (C * H * W);   // 128 for the reference shapes
    const int planes = nBatch * C;                  // 384
    const dim3 grid(planes * TILES);                // 5376 blocks
    const dim3 block(W);                            // 224 threads = 7 wave32

    sobel_gabor_fused_kernel<<<grid, block, 0, stream>>>(x, mask, out, nBatch);
}